// LastAggregator_24077586661672
// MI455X (gfx1250) — compile-verified
//
#include <hip/hip_runtime.h>
#include <stdint.h>

// ---------------------------------------------------------------------------
// LastAggregator: segmented argmax-by-timestamp + row gather.
// Single-pass argmax via 64-bit packed keys and GLOBAL_ATOMIC_MAX_U64:
//   key = (ordered(t_bits) << 32) | ~edge_id
// max over keys == max t; ties -> smallest edge id (first occurrence),
// exactly matching the reference's segment_max + first-position argmax.
// ---------------------------------------------------------------------------

typedef float v4f __attribute__((ext_vector_type(4)));   // true vector type for NT builtins

__device__ __forceinline__ uint32_t ordered_f32(float f) {
    // Monotone map float -> uint32 (handles negatives; t here is [0,1)).
    uint32_t b = __float_as_uint(f);
    return (b & 0x80000000u) ? ~b : (b | 0x80000000u);
}

__global__ void la_init_keys(unsigned long long* __restrict__ keys, int n) {
    int i = blockIdx.x * blockDim.x + threadIdx.x;
    if (i < n) keys[i] = 0ull;   // 0 == "empty segment" sentinel (real keys have low word != 0)
}

__global__ void la_scatter_argmax(const int* __restrict__ index,
                                  const float* __restrict__ t,
                                  unsigned long long* __restrict__ keys,
                                  int num_edges) {
    int e = blockIdx.x * blockDim.x + threadIdx.x;
    if (e >= num_edges) return;
    // read-once streams -> non-temporal loads (gfx1250 TH_NT)
    int   idx = __builtin_nontemporal_load(index + e);
    float tv  = __builtin_nontemporal_load(t + e);
    uint32_t hi = ordered_f32(tv);
    uint32_t lo = 0xFFFFFFFFu - (uint32_t)e;             // ~edge_id: bigger == earlier edge
    unsigned long long key = ((unsigned long long)hi << 32) | (unsigned long long)lo;
    // 800KB key table << 192MB L2: RMW stays on-chip. -> global_atomic_max_u64
    atomicMax(&keys[idx], key);
}

// D4 = D/4 (float4 chunks per row). Compile-time specialization avoids the
// 64-bit div/mod VALU expansion; D4==32 -> pure shifts, one wave per row.
template <int D4>
__global__ void la_gather_rows_t(const unsigned long long* __restrict__ keys,
                                 const float* __restrict__ msg,
                                 float* __restrict__ out,
                                 int n_nodes) {
    long long tid = (long long)blockIdx.x * blockDim.x + threadIdx.x;
    int node  = (int)(tid / D4);
    int chunk = (int)(tid % D4);
    if (node >= n_nodes) return;

    unsigned long long key = keys[node];   // wave-uniform (D4==32): one L2 request
    v4f v = (v4f)0.f;
    if (key != 0ull) {
        uint32_t edge = 0xFFFFFFFFu - (uint32_t)(key & 0xFFFFFFFFull);
        const v4f* __restrict__ src = (const v4f*)(msg + (size_t)edge * (size_t)(D4 * 4));
        v = __builtin_nontemporal_load(src + chunk);     // global_load_b128, read-once row
    }
    v4f* __restrict__ dst = (v4f*)(out + (size_t)node * (size_t)(D4 * 4));
    __builtin_nontemporal_store(v, dst + chunk);         // global_store_b128, write-once
}

// Generic fallback for unexpected D (runtime d4).
__global__ void la_gather_rows_g(const unsigned long long* __restrict__ keys,
                                 const float* __restrict__ msg,
                                 float* __restrict__ out,
                                 int n_nodes, int d4) {
    long long tid = (long long)blockIdx.x * blockDim.x + threadIdx.x;
    int node  = (int)(tid / d4);
    int chunk = (int)(tid % d4);
    if (node >= n_nodes) return;

    unsigned long long key = keys[node];
    v4f v = (v4f)0.f;
    if (key != 0ull) {
        uint32_t edge = 0xFFFFFFFFu - (uint32_t)(key & 0xFFFFFFFFull);
        const v4f* __restrict__ src = (const v4f*)(msg + (size_t)edge * (size_t)(d4 * 4));
        v = __builtin_nontemporal_load(src + chunk);
    }
    v4f* __restrict__ dst = (v4f*)(out + (size_t)node * (size_t)(d4 * 4));
    __builtin_nontemporal_store(v, dst + chunk);
}

extern "C" void kernel_launch(void* const* d_in, const int* in_sizes, int n_in,
                              void* d_out, int out_size, void* d_ws, size_t ws_size,
                              hipStream_t stream) {
    // Input order per setup_inputs(): msg[E*D] f32, index[E] i32, t[E] f32, dim_size (scalar)
    const float* msg   = (const float*)d_in[0];
    const int*   index = (const int*)  d_in[1];
    const float* t     = (const float*)d_in[2];
    float*       out   = (float*)d_out;

    const int E = in_sizes[1];
    const int D = in_sizes[0] / E;       // 128
    const int N = out_size / D;          // 100000 (dim_size lives on device; recover from out_size)
    const int d4 = D / 4;                // float4 chunks per row (32)

    unsigned long long* keys = (unsigned long long*)d_ws;   // N * 8 bytes of scratch

    const int threads = 256;             // 8 waves per block on wave32 CDNA5

    // 1) zero key table (every call: harness doesn't re-poison d_ws between replays)
    la_init_keys<<<(N + threads - 1) / threads, threads, 0, stream>>>(keys, N);

    // 2) one u64 atomicMax per edge (L2-resident)
    la_scatter_argmax<<<(E + threads - 1) / threads, threads, 0, stream>>>(index, t, keys, E);

    // 3) decode winners, stream rows at B128 granularity
    long long total = (long long)N * (long long)d4;
    int blocks = (int)((total + threads - 1) / threads);
    if (d4 == 32) {
        la_gather_rows_t<32><<<blocks, threads, 0, stream>>>(keys, msg, out, N);
    } else {
        la_gather_rows_g<<<blocks, threads, 0, stream>>>(keys, msg, out, N, d4);
    }
}